// DECKS_13649406067409
// MI455X (gfx1250) — compile-verified
//
#include <hip/hip_runtime.h>

// ---------------------------------------------------------------------------
// CDNA5 / gfx1250 implementation.
//  Phase 1: per-wave V_WMMA_F32_16X16X4_F32 computes [x@points | rowsum(x)]
//           for 16-row tiles (K=6 padded to 8 via two accumulating WMMAs;
//           B column 3 is all-ones so the row sum falls out of the same op).
//           Results land in LDS as float4 (wx, wy, wz, sum) per row.
//  Phase 2: each thread owns SEG=15 rows and replays the linear recurrence
//           dyn = 0.8*dyn + 0.2*wmp from 192 rows earlier (0.8^192 ~ 4e-19,
//           below fp32 resolution; exact for block 0), then emits the 6
//           normalized anchor distances.
// ---------------------------------------------------------------------------

typedef __attribute__((ext_vector_type(2))) float v2f;
typedef __attribute__((ext_vector_type(8))) float v8f;

#define CHUNK   3840              // output rows per block (= 256 threads * SEG)
#define SEG     15                // output rows per thread
#define WARMUP  192               // scan warm-up depth (0.8^192 ~ 4e-19)
#define RROWS   (CHUNK + WARMUP)  // rows staged in LDS = 4032
#define NTILES  (RROWS / 16)      // 252 WMMA tiles per block
#define SHIFT   0.2f
#define MAX_D   1.7399137878f     // max pairwise anchor distance (P4<->P3)

// B matrix for D = A*B, stored row-major [k][n]:
//   columns 0..2 = anchor-point coords, column 3 = 1.0 (row-sum lane),
//   rows 6,7 are zero padding for the K=8 second WMMA.
__device__ __constant__ float BTAB[8][16] = {
    {-0.51f,  0.59f,  0.25f, 1.0f},
    {-0.64f,  0.60f, -0.43f, 1.0f},
    { 0.76f,  0.48f,  0.35f, 1.0f},
    { 0.82f,  0.65f, -0.05f, 1.0f},
    {-0.63f, -0.27f, -0.33f, 1.0f},
    { 0.40f,  0.67f, -0.13f, 1.0f},
    { 0.0f },
    { 0.0f },
};

__global__ void DECKS_13649406067409_kernel(const float* __restrict__ x,
                                            float* __restrict__ out,
                                            int N) {
    extern __shared__ float s_wmp[];          // RROWS * 4 floats (wx,wy,wz,sum)

    const int g0   = blockIdx.x * CHUNK;      // first output row of this block
    const int base = g0 - WARMUP;             // global row of LDS row 0
    const int lane = threadIdx.x & 31;        // wave32
    const int wave = threadIdx.x >> 5;        // 8 waves per block

    // ---- Phase 1: WMMA [16x8]x[8x16] tiles -> (x@P | rowsum) into LDS ----
    const int  n  = lane & 15;
    const bool hi = lane >= 16;

    // B fragment (32-bit 4x16): v0 holds K = hi?2:0, v1 holds K+1, N = lane&15.
    v2f b0, b1;
    b0.x = BTAB[hi ? 2 : 0][n];
    b0.y = BTAB[hi ? 3 : 1][n];
    b1.x = BTAB[hi ? 6 : 4][n];   // rows 6,7 are zero padding
    b1.y = BTAB[hi ? 7 : 5][n];

    for (int tile = wave; tile < NTILES; tile += 8) {
        int lrow = tile * 16 + n;                 // LDS-local row this lane owns
        int grow = base + lrow;
        grow = min(max(grow, 0), N - 1);          // clamp (block 0 / tail)
        const float* xr = x + (size_t)grow * 6;

        // A fragment (32-bit 16x4): lanes 0-15 hold K=0,1; lanes 16-31 K=2,3.
        v2f a0, a1;
        if (!hi) {
            float2 lo = *(const float2*)(xr + 0); // k=0,1
            float2 h4 = *(const float2*)(xr + 4); // k=4,5 (second WMMA)
            a0.x = lo.x; a0.y = lo.y;
            a1.x = h4.x; a1.y = h4.y;
        } else {
            float2 mid = *(const float2*)(xr + 2); // k=2,3
            a0.x = mid.x; a0.y = mid.y;
            a1.x = 0.0f;  a1.y = 0.0f;             // k=6,7 padding
        }

        // Prefetch next tile's row while the WMMAs run.
        if (tile + 8 < NTILES) {
            int pg = min(max(base + (tile + 8) * 16 + n, 0), N - 1);
            __builtin_prefetch(x + (size_t)pg * 6, 0, 1);
        }

        v8f c = {};
        c = __builtin_amdgcn_wmma_f32_16x16x4_f32(false, a0, false, b0,
                                                  (short)0, c, false, false);
        c = __builtin_amdgcn_wmma_f32_16x16x4_f32(false, a1, false, b1,
                                                  (short)0, c, false, false);

        // D layout: VGPR v holds M=v (lanes 0-15) / M=8+v (lanes 16-31), N=lane&15.
        // Columns 0..2 = weighted coords, column 3 = row sum.
        if (n < 4) {
            int mofs = hi ? 8 : 0;
#pragma unroll
            for (int v = 0; v < 8; ++v)
                s_wmp[(tile * 16 + mofs + v) * 4 + n] = c[v];
        }
    }
    __syncthreads();

    // ---- Phase 2: warm-up linear scan + distance emit ----
    const float4* wm = (const float4*)s_wmp;

    const int l0   = WARMUP + (int)threadIdx.x * SEG; // first owned LDS row
    int lws        = l0 - WARMUP;                     // warm-up start
    const int clmp = WARMUP - g0;                     // first LDS row with grow>=0
    if (lws < clmp) lws = clmp;                       // (only block 0)

    float4 w0 = wm[lws];
    float  rs = __builtin_amdgcn_rcpf(w0.w);
    float  dx = w0.x * rs, dy = w0.y * rs, dz = w0.z * rs;

    const float PP[6][3] = {
        {-0.51f,  0.59f,  0.25f}, {-0.64f,  0.60f, -0.43f},
        { 0.76f,  0.48f,  0.35f}, { 0.82f,  0.65f, -0.05f},
        {-0.63f, -0.27f, -0.33f}, { 0.40f,  0.67f, -0.13f}};
    const float INV_MAXD = 1.0f / MAX_D;

    auto emit = [&](int l, float ex, float ey, float ez) {
        int grow = base + l;
        if (grow >= N) return;
        float res[6];
#pragma unroll
        for (int p = 0; p < 6; ++p) {
            float ddx = ex - PP[p][0];
            float ddy = ey - PP[p][1];
            float ddz = ez - PP[p][2];
            float d2  = __builtin_fmaf(ddx, ddx,
                        __builtin_fmaf(ddy, ddy, ddz * ddz));
            res[p] = __builtin_fmaf(-__builtin_sqrtf(d2), INV_MAXD, 1.0f);
        }
        float2* o2 = (float2*)(out + (size_t)grow * 6);
        o2[0] = make_float2(res[0], res[1]);
        o2[1] = make_float2(res[2], res[3]);
        o2[2] = make_float2(res[4], res[5]);
    };

    if (lws >= l0) emit(lws, dx, dy, dz);   // row 0 of the grid (block 0, t=0)

    const int lend = l0 + SEG;
    for (int l = lws + 1; l < lend; ++l) {
        float4 w = wm[l];
        float  r  = __builtin_amdgcn_rcpf(w.w);
        float  wx = w.x * r, wy = w.y * r, wz = w.z * r;
        dx -= (dx - wx) * SHIFT;
        dy -= (dy - wy) * SHIFT;
        dz -= (dz - wz) * SHIFT;
        if (l >= l0) emit(l, dx, dy, dz);
    }
}

extern "C" void kernel_launch(void* const* d_in, const int* in_sizes, int n_in,
                              void* d_out, int out_size, void* d_ws, size_t ws_size,
                              hipStream_t stream) {
    const float* x   = (const float*)d_in[0];
    float*       out = (float*)d_out;
    const int    N   = in_sizes[0] / 6;      // x is [N, 6]

    const int    blocks = (N + CHUNK - 1) / CHUNK;
    const size_t shmem  = (size_t)RROWS * 4 * sizeof(float);   // 63 KB

    DECKS_13649406067409_kernel<<<dim3(blocks), dim3(256), shmem, stream>>>(x, out, N);
}